// RelationNetwork_996432413428
// MI455X (gfx1250) — compile-verified
//
#include <hip/hip_runtime.h>

typedef __attribute__((ext_vector_type(2))) float v2f;
typedef __attribute__((ext_vector_type(8))) float v8f;

#define B_    8
#define N_    20
#define K_    5
#define Q_    15
#define D_    1024
#define NQ_   300    // N_*Q_
#define JD_   20     // K_+Q_
#define DC_   256    // D chunk staged in LDS
#define PITCH_ 260   // DC_+4 dwords: kills 64-bank conflicts on row-parallel reads
#define NUM_QT_ 19   // ceil(300/16)

// ---------------- Kernel 1: proto = mean_k(support), term_p = proto . w0 -------------
__global__ void __launch_bounds__(256)
k_proto(const float* __restrict__ emb, const float* __restrict__ w,
        float* __restrict__ proto, float* __restrict__ term_p) {
  int bn = blockIdx.x;            // 0..159  (b*N + n)
  const float* base = emb + (size_t)bn * JD_ * D_;
  float partial = 0.f;
  for (int d = threadIdx.x; d < D_; d += 256) {
    float s = 0.f;
    #pragma unroll
    for (int k = 0; k < K_; ++k) s += base[k * D_ + d];
    s *= 0.2f;                    // mean over K=5
    proto[(size_t)bn * D_ + d] = s;
    partial += s * w[d];          // w0
  }
  __shared__ float red[256];
  red[threadIdx.x] = partial;
  __syncthreads();
  for (int off = 128; off > 0; off >>= 1) {
    if ((int)threadIdx.x < off) red[threadIdx.x] += red[threadIdx.x + off];
    __syncthreads();
  }
  if (threadIdx.x == 0) term_p[bn] = red[0];
}

// ---------------- Kernel 2: term_q = query . w1 (one wave32 per query) ---------------
__global__ void __launch_bounds__(256)
k_termq(const float* __restrict__ emb, const float* __restrict__ w,
        float* __restrict__ term_q) {
  int wave = blockIdx.x * 8 + (threadIdx.x >> 5);
  int lane = threadIdx.x & 31;
  if (wave >= B_ * NQ_) return;
  int b = wave / NQ_, qq = wave % NQ_;
  int n = qq / Q_, qi = qq % Q_;
  const float* qrow = emb + ((size_t)((b * N_ + n) * JD_ + K_ + qi)) * D_;
  const float* w1 = w + D_;
  float s = 0.f;
  for (int d = lane; d < D_; d += 32) s += qrow[d] * w1[d];
  #pragma unroll
  for (int off = 16; off > 0; off >>= 1) s += __shfl_down(s, off, 32);
  if (lane == 0) term_q[wave] = s;
}

// ---------------- Kernel 3: fused abs-term (VALU) + cos-term (WMMA f32) --------------
__global__ void __launch_bounds__(64)
k_main(const float* __restrict__ emb, const float* __restrict__ w,
       const float* __restrict__ bias, const float* __restrict__ proto,
       const float* __restrict__ term_p, const float* __restrict__ term_q,
       float* __restrict__ out) {
  __shared__ float s_q[16 * PITCH_];   // 16 query rows, one D-chunk
  __shared__ float s_p[32 * PITCH_];   // 32 proto rows (20 valid, rest zero)
  __shared__ float s_w2[DC_];
  __shared__ float s_w3[DC_];

  const int qt   = blockIdx.x;         // 0..18 q-tile
  const int b    = blockIdx.y;         // 0..7
  const int tid  = threadIdx.x;        // 0..63
  const int wv   = tid >> 5;           // wave = n-tile (0: n 0..15, 1: n 16..31)
  const int lane = tid & 31;
  const int hi   = lane >> 4;          // lane >= 16
  const int lm   = lane & 15;
  const int ncol = wv * 16 + lm;       // output column n this lane owns (0..31)
  const int q0   = qt * 16;

  v8f   acc = {};                      // WMMA C/D: cos term
  float aacc[8] = {0.f,0.f,0.f,0.f,0.f,0.f,0.f,0.f};  // abs term, same (m,n) layout

  for (int c = 0; c < D_ / DC_; ++c) {
    const int dbase = c * DC_;

    // ---- stage query tile (rows >= NQ_ zero-padded) ----
    for (int idx = tid; idx < 16 * (DC_ / 4); idx += 64) {
      int row = idx / (DC_ / 4);
      int c4  = idx % (DC_ / 4);
      float4 v = make_float4(0.f, 0.f, 0.f, 0.f);
      int qq = q0 + row;
      if (qq < NQ_) {
        int n = qq / Q_, qi = qq % Q_;
        const float4* src = (const float4*)(emb +
            ((size_t)((b * N_ + n) * JD_ + K_ + qi)) * D_ + dbase);
        v = src[c4];
      }
      *(float4*)&s_q[row * PITCH_ + c4 * 4] = v;
    }
    // ---- stage proto rows (rows >= N_ zero-padded) ----
    for (int idx = tid; idx < 32 * (DC_ / 4); idx += 64) {
      int row = idx / (DC_ / 4);
      int c4  = idx % (DC_ / 4);
      float4 v = make_float4(0.f, 0.f, 0.f, 0.f);
      if (row < N_) {
        const float4* src = (const float4*)(proto + (size_t)(b * N_ + row) * D_ + dbase);
        v = src[c4];
      }
      *(float4*)&s_p[row * PITCH_ + c4 * 4] = v;
    }
    // ---- stage w2 / w3 chunks ----
    for (int idx = tid; idx < DC_ / 4; idx += 64) {
      *(float4*)&s_w2[idx * 4] = ((const float4*)(w + 2 * D_ + dbase))[idx];
      *(float4*)&s_w3[idx * 4] = ((const float4*)(w + 3 * D_ + dbase))[idx];
    }
    __syncthreads();

    // ---- cos term: V_WMMA_F32_16X16X4_F32, K-steps of 4 over the chunk ----
    // A(16x4) lane layout: row=lane%16, v0=K(k0+2*hi), v1=K(+1).  A = query * w3.
    // B(4x16)  lane layout: col=lane%16, v0=row K(k0+2*hi), v1=K(+1). B = proto^T.
    #pragma unroll 4
    for (int kk = 0; kk < DC_; kk += 4) {
      int kf = kk + 2 * hi;
      v2f a, bb;
      a.x  = s_q[lm * PITCH_ + kf]     * s_w3[kf];
      a.y  = s_q[lm * PITCH_ + kf + 1] * s_w3[kf + 1];
      bb.x = s_p[ncol * PITCH_ + kf];
      bb.y = s_p[ncol * PITCH_ + kf + 1];
      acc = __builtin_amdgcn_wmma_f32_16x16x4_f32(
          /*neg_a=*/false, a, /*neg_b=*/false, bb,
          /*c_mod=*/(short)0, acc, /*reuse_a=*/false, /*reuse_b=*/false);
    }

    // ---- abs term: aacc[r] += sum_d |proto[ncol,d] - query[m,d]| * w2[d] ----
    for (int d4 = 0; d4 < DC_; d4 += 4) {
      float4 pv = *(const float4*)&s_p[ncol * PITCH_ + d4];
      float4 wvv = *(const float4*)&s_w2[d4];
      #pragma unroll
      for (int r = 0; r < 8; ++r) {
        int m = r + 8 * hi;            // matches WMMA C row mapping
        float4 qv = *(const float4*)&s_q[m * PITCH_ + d4];
        aacc[r] += fabsf(pv.x - qv.x) * wvv.x + fabsf(pv.y - qv.y) * wvv.y +
                   fabsf(pv.z - qv.z) * wvv.z + fabsf(pv.w - qv.w) * wvv.w;
      }
    }
    __syncthreads();
  }

  // ---- epilogue: combine all four terms + bias ----
  float bs = bias[0];
  float tp = (ncol < N_) ? term_p[b * N_ + ncol] : 0.f;
  #pragma unroll
  for (int r = 0; r < 8; ++r) {
    int m  = r + 8 * hi;
    int qq = q0 + m;
    if (qq < NQ_ && ncol < N_) {
      float val = acc[r] + aacc[r] + tp + term_q[b * NQ_ + qq] + bs;
      out[(size_t)b * NQ_ * N_ + (size_t)qq * N_ + ncol] = val;
    }
  }
}

// -------------------------------- launcher -------------------------------------------
extern "C" void kernel_launch(void* const* d_in, const int* in_sizes, int n_in,
                              void* d_out, int out_size, void* d_ws, size_t ws_size,
                              hipStream_t stream) {
  const float* emb  = (const float*)d_in[0];
  const float* w    = (const float*)d_in[1];
  const float* bias = (const float*)d_in[2];
  float* out = (float*)d_out;

  float* ws     = (float*)d_ws;
  float* proto  = ws;                          // B*N*D = 163840 floats
  float* term_p = proto + (size_t)B_ * N_ * D_; // 160 floats
  float* term_q = term_p + B_ * N_;             // 2400 floats

  k_proto<<<dim3(B_ * N_), dim3(256), 0, stream>>>(emb, w, proto, term_p);
  k_termq<<<dim3((B_ * NQ_) / 8), dim3(256), 0, stream>>>(emb, w, term_q);
  k_main <<<dim3(NUM_QT_, B_), dim3(64), 0, stream>>>(emb, w, bias, proto,
                                                      term_p, term_q, out);
}